// VGAE_55903294324759
// MI455X (gfx1250) — compile-verified
//
#include <hip/hip_runtime.h>

typedef __attribute__((ext_vector_type(2))) float v2f;
typedef __attribute__((ext_vector_type(8))) float v8f;

#define N_NODES 8192
#define N_EDGES 262144
#define IN_DIM  512
#define HID     256
#define EMB     64

__device__ __forceinline__ v8f wmma_f32(v2f a, v2f b, v8f c) {
    return __builtin_amdgcn_wmma_f32_16x16x4_f32(false, a, false, b,
                                                 (short)0, c, false, false);
}

// ---------------------------------------------------------------------------
// fp32 WMMA GEMM, compile-time N/K: C[M,N] = A[M,K] @ B[K,N], row-major.
// One wave -> one 16x16 C tile; full K unroll -> all loads are base+imm offset.
// V_WMMA_F32_16X16X4_F32 fragment layout (ISA 7.12.2, 32-bit A 16x4):
//   A: lanes 0-15 hold M=lane, VGPR pair = K={k0,k0+1}; lanes 16-31 = K={k0+2,k0+3}
//   B (4x16, mirrored): lanes hold N, VGPR pair holds the K pair
//   C/D: vgpr r, lanes 0-15 -> M=r, N=lane; lanes 16-31 -> M=r+8
// ---------------------------------------------------------------------------
template<int Nn, int K>
__global__ void gemm_f32_wmma(const float* __restrict__ A, const float* __restrict__ B,
                              float* __restrict__ C) {
    constexpr int ntn = Nn / 16;
    const int wid  = threadIdx.x >> 5;
    const int lane = threadIdx.x & 31;
    const int half = lane >> 4;
    const int l    = lane & 15;
    const int tile = blockIdx.x * 8 + wid;     // grid sized so tiles % 8 == 0
    const int tm = tile / ntn;
    const int tn = tile - tm * ntn;

    // Hoist the half-wave K offset into the bases; all loop offsets are constants.
    const float* Arow  = A + (size_t)(tm * 16 + l) * K + 2 * half;        // 8B aligned
    const float* Bbase = B + (size_t)(2 * half) * Nn + tn * 16 + l;

    v8f acc = {};
#pragma unroll
    for (int k0 = 0; k0 < K; k0 += 4) {
        v2f a = *(const v2f*)(Arow + k0);                       // global_load_b64
        v2f b = { Bbase[k0 * Nn], Bbase[k0 * Nn + Nn] };        // 2x b32, imm offsets
        acc = wmma_f32(a, b, acc);
    }

    float* Crow = C + (size_t)(tm * 16 + half * 8) * Nn + tn * 16 + l;
#pragma unroll
    for (int r = 0; r < 8; ++r)
        Crow[(size_t)r * Nn] = acc[r];
}

// ---------------------------------------------------------------------------
// Decoder: out[8192,8192] = Z @ Z^T, Z = [8192, EMB] fp32 (2 MB, L2-resident).
// B = Z^T => B[k][n] = Z[n][k]: B fragments gather rows of Z like A fragments.
// Each wave computes a 16x64 strip: 4 independent accumulators share every A
// fragment (1.25 loads/wmma) and break the WMMA->WMMA RAW dependency chain.
// 268 MB output written with non-temporal stores (bypasses 192 MB L2).
// grid = (512/(8*4), 512), block = 256 (8 waves).
// ---------------------------------------------------------------------------
__global__ void zzt_wmma(const float* __restrict__ Z, float* __restrict__ out) {
    const int wid  = threadIdx.x >> 5;
    const int lane = threadIdx.x & 31;
    const int half = lane >> 4;
    const int l    = lane & 15;
    const int tn0 = (blockIdx.x * 8 + wid) * 4;   // first of 4 adjacent n-tiles
    const int tm  = blockIdx.y;

    const float* Arow = Z + (size_t)(tm * 16 + l) * EMB + 2 * half;
    const float* B0   = Z + (size_t)(tn0 * 16 + l) * EMB + 2 * half;

    v8f acc0 = {}, acc1 = {}, acc2 = {}, acc3 = {};
#pragma unroll
    for (int k0 = 0; k0 < EMB; k0 += 4) {
        v2f a  = *(const v2f*)(Arow + k0);
        v2f b0 = *(const v2f*)(B0 + k0);
        v2f b1 = *(const v2f*)(B0 + 16 * EMB + k0);
        v2f b2 = *(const v2f*)(B0 + 32 * EMB + k0);
        v2f b3 = *(const v2f*)(B0 + 48 * EMB + k0);
        acc0 = wmma_f32(a, b0, acc0);
        acc1 = wmma_f32(a, b1, acc1);
        acc2 = wmma_f32(a, b2, acc2);
        acc3 = wmma_f32(a, b3, acc3);
    }

    float* Crow = out + (size_t)(tm * 16 + half * 8) * N_NODES + tn0 * 16 + l;
#pragma unroll
    for (int r = 0; r < 8; ++r) {
        float* p = Crow + (size_t)r * N_NODES;
        __builtin_nontemporal_store(acc0[r], p);
        __builtin_nontemporal_store(acc1[r], p + 16);
        __builtin_nontemporal_store(acc2[r], p + 32);
        __builtin_nontemporal_store(acc3[r], p + 48);
    }
}

// ---------------------------------------------------------------------------
// Edge scatter-add: out[dst[e], :] += H[src[e], :] * w[e]
// One block per edge; hardware fp32 atomics (GLOBAL_ATOMIC_ADD_F32, no return).
// ---------------------------------------------------------------------------
__global__ void scatter_add_edges(const float* __restrict__ Hm, const int* __restrict__ src,
                                  const int* __restrict__ dst, const float* __restrict__ w,
                                  float* __restrict__ out, int D) {
    const int e = blockIdx.x;
    const int s = src[e];
    const int d = dst[e];
    const float we = w[e];
    const float* hrow = Hm + (size_t)s * D;
    float* orow = out + (size_t)d * D;
    for (int t = threadIdx.x; t < D; t += blockDim.x)
        unsafeAtomicAdd(&orow[t], hrow[t] * we);
}

// ---------------------------------------------------------------------------
// Reparameterization: Z = noise * exp(relu(logstd)) + relu(mean)
// ---------------------------------------------------------------------------
__global__ void reparam(const float* __restrict__ meanA, const float* __restrict__ logsA,
                        const float* __restrict__ noise, float* __restrict__ Z, int n) {
    const int i = blockIdx.x * blockDim.x + threadIdx.x;
    if (i < n) {
        const float m  = fmaxf(meanA[i], 0.0f);
        const float ls = fmaxf(logsA[i], 0.0f);
        Z[i] = noise[i] * expf(ls) + m;
    }
}

extern "C" void kernel_launch(void* const* d_in, const int* in_sizes, int n_in,
                              void* d_out, int out_size, void* d_ws, size_t ws_size,
                              hipStream_t stream) {
    const float* features    = (const float*)d_in[0];
    const int*   edge_index  = (const int*)  d_in[1];
    const float* edge_weight = (const float*)d_in[2];
    const float* W0          = (const float*)d_in[3];
    const float* W_mean      = (const float*)d_in[4];
    const float* W_logstd    = (const float*)d_in[5];
    const float* noise       = (const float*)d_in[6];
    float* out = (float*)d_out;

    const int* src = edge_index;            // edge_index[0, :]
    const int* dst = edge_index + N_EDGES;  // edge_index[1, :]

    // Workspace layout (fp32): total ~27.3 MB
    float* H      = (float*)d_ws;                       // [N, HID]
    float* hidden = H      + (size_t)N_NODES * HID;     // [N, HID] (accumulator)
    float* M0     = hidden + (size_t)N_NODES * HID;     // [N, EMB]
    float* L0     = M0     + (size_t)N_NODES * EMB;     // [N, EMB]
    float* meanA  = L0     + (size_t)N_NODES * EMB;     // [N, EMB] (accumulator)
    float* logsA  = meanA  + (size_t)N_NODES * EMB;     // [N, EMB] (accumulator)
    float* Z      = logsA  + (size_t)N_NODES * EMB;     // [N, EMB]

    // Zero the scatter accumulators every call (deterministic, capture-safe).
    hipMemsetAsync(hidden, 0, sizeof(float) * (size_t)N_NODES * HID, stream);
    hipMemsetAsync(meanA,  0, sizeof(float) * (size_t)N_NODES * EMB, stream);
    hipMemsetAsync(logsA,  0, sizeof(float) * (size_t)N_NODES * EMB, stream);

    // Stage A: H = features @ W0        (8192x512 @ 512x256) -> 8192 tiles
    gemm_f32_wmma<HID, IN_DIM>
        <<<(N_NODES / 16) * (HID / 16) / 8, 256, 0, stream>>>(features, W0, H);

    // Stage B: hidden = scatter_add(H[src] * w -> dst)
    scatter_add_edges<<<N_EDGES, 256, 0, stream>>>(H, src, dst, edge_weight, hidden, HID);

    // Stage C: M0 = hidden @ W_mean ; L0 = hidden @ W_logstd  (8192x256 @ 256x64)
    gemm_f32_wmma<EMB, HID>
        <<<(N_NODES / 16) * (EMB / 16) / 8, 256, 0, stream>>>(hidden, W_mean, M0);
    gemm_f32_wmma<EMB, HID>
        <<<(N_NODES / 16) * (EMB / 16) / 8, 256, 0, stream>>>(hidden, W_logstd, L0);

    scatter_add_edges<<<N_EDGES, 64, 0, stream>>>(M0, src, dst, edge_weight, meanA, EMB);
    scatter_add_edges<<<N_EDGES, 64, 0, stream>>>(L0, src, dst, edge_weight, logsA, EMB);

    // Stage D: Z = noise * exp(relu(logstd)) + relu(mean)
    {
        const int nz = N_NODES * EMB;
        reparam<<<(nz + 255) / 256, 256, 0, stream>>>(meanA, logsA, noise, Z, nz);
    }

    // Stage E: adj_logits = Z @ Z^T  (268 MB NT stores; HBM-write bound)
    {
        dim3 grid(N_NODES / 16 / 8 / 4, N_NODES / 16);
        zzt_wmma<<<grid, 256, 0, stream>>>(Z, out);
    }
}